// LSTM_with_Attention_67018669687291
// MI455X (gfx1250) — compile-verified
//
#include <hip/hip_runtime.h>
#include <hip/hip_bf16.h>
#include <math.h>

typedef __attribute__((ext_vector_type(2))) float v2f;
typedef __attribute__((ext_vector_type(8))) float v8f;

#define BB   8
#define SS   2048
#define INF_ 64
#define HH   128
#define G4   512
#define OUTD 64

__device__ __forceinline__ float sigm(float x) { return 1.0f / (1.0f + __expf(-x)); }

// ---------------------------------------------------------------------------
// Kernel 1: x_proj[b,s,4H] = x[b,s,:] @ W_ih^T + (b_ih + b_hh)
// GEMM [B*S=16384, 64] x [64, 512]. One wave per 16x16 output tile.
// grid = (1024, 8), block = 128 (4 waves); wave w handles N-tile blockIdx.y*4+w.
// ---------------------------------------------------------------------------
__global__ void xproj_kernel(const float* __restrict__ x,
                             const float* __restrict__ Wih,
                             const float* __restrict__ bih,
                             const float* __restrict__ bhh,
                             float* __restrict__ xproj) {
  const int lane = threadIdx.x & 31;
  const int wave = threadIdx.x >> 5;
  const int mt   = blockIdx.x;                 // 0..1023  (rows of [B*S])
  const int nt   = blockIdx.y * 4 + wave;      // 0..31    (cols of 4H)
  const int col  = lane & 15;
  const int hi   = lane >> 4;                  // 0 or 1
  const int n    = nt * 16 + col;

  const float bias = bih[n] + bhh[n];
  v8f acc;
#pragma unroll
  for (int v = 0; v < 8; ++v) acc[v] = bias;

  const float* xrow = x   + (size_t)(mt * 16 + col) * INF_;  // A row = lane&15
  const float* wrow = Wih + (size_t)n * INF_;                // B col = lane&15
#pragma unroll
  for (int ks = 0; ks < INF_ / 4; ++ks) {
    const int k = ks * 4 + hi * 2;
    v2f a, b;
    a.x = xrow[k]; a.y = xrow[k + 1];
    b.x = wrow[k]; b.y = wrow[k + 1];
    acc = __builtin_amdgcn_wmma_f32_16x16x4_f32(false, a, false, b,
                                                (short)0, acc, false, false);
  }
#pragma unroll
  for (int v = 0; v < 8; ++v)
    xproj[(size_t)(mt * 16 + v + 8 * hi) * G4 + n] = acc[v];
}

// ---------------------------------------------------------------------------
// Kernel 2: sequential LSTM scan. ONE persistent workgroup of 1024 threads
// (32 waves). Wave w owns gate-columns [16w,16w+16); its W_hh B-fragments are
// loop-invariant and live in registers (64 VGPRs). h/c/gates live in LDS.
// Per step: 32 waves x 32 WMMA(16x16x4) => gates[16,512]; then 1024 threads
// apply the pointwise cell update. Two barriers per step.
// ---------------------------------------------------------------------------
__global__ void __launch_bounds__(1024)
lstm_kernel(const float* __restrict__ xproj,
            const float* __restrict__ h0, const float* __restrict__ c0,
            const float* __restrict__ Whh,
            float* __restrict__ lout,            // [B,S,H]
            float* __restrict__ hn, float* __restrict__ cn) {
  __shared__ float h_lds[16][HH];
  __shared__ float c_lds[BB][HH];
  __shared__ float gates[16][G4];

  const int tid  = threadIdx.x;
  const int lane = tid & 31;
  const int nt   = tid >> 5;                   // wave id == N-tile, 0..31
  const int col  = lane & 15;
  const int hi   = lane >> 4;

  {  // init recurrent state (pad rows 8..15 of h to zero)
    const int b0 = tid >> 7, j = tid & 127;
    h_lds[b0][j]     = h0[tid];
    h_lds[b0 + 8][j] = 0.0f;
    c_lds[b0][j]     = c0[tid];
  }

  // Preload W_hh fragments: B[k][n] = Whh[(nt*16+n)*H + k]
  v2f wf[32];
  {
    const float* wrow = Whh + (size_t)(nt * 16 + col) * HH;
#pragma unroll
    for (int ks = 0; ks < 32; ++ks) {
      const int k = ks * 4 + hi * 2;
      wf[ks].x = wrow[k]; wf[ks].y = wrow[k + 1];
    }
  }
  __syncthreads();

  for (int t = 0; t < SS; ++t) {
    // --- phase 1: gates tile = x_proj[:,t,:] + h @ W_hh^T ------------------
    v8f acc;
#pragma unroll
    for (int v = 0; v < 8; ++v) {
      const int m = (v + 8 * hi) & 7;  // pad rows alias valid batches (discarded)
      acc[v] = xproj[((size_t)m * SS + t) * G4 + nt * 16 + col];
    }
#pragma unroll
    for (int ks = 0; ks < 32; ++ks) {
      const int k = ks * 4 + hi * 2;
      v2f a;
      a.x = h_lds[col][k]; a.y = h_lds[col][k + 1];
      acc = __builtin_amdgcn_wmma_f32_16x16x4_f32(false, a, false, wf[ks],
                                                  (short)0, acc, false, false);
    }
#pragma unroll
    for (int v = 0; v < 8; ++v)
      gates[v + 8 * hi][nt * 16 + col] = acc[v];
    __syncthreads();

    // --- phase 2: pointwise cell update (gate order: i, f, g, o) -----------
    {
      const int b0 = tid >> 7, j = tid & 127;
      const float gi = gates[b0][j];
      const float gf = gates[b0][HH + j];
      const float gg = gates[b0][2 * HH + j];
      const float go = gates[b0][3 * HH + j];
      const float cc = sigm(gf) * c_lds[b0][j] + sigm(gi) * tanhf(gg);
      const float hh = sigm(go) * tanhf(cc);
      c_lds[b0][j] = cc;
      h_lds[b0][j] = hh;
      lout[((size_t)b0 * SS + t) * HH + j] = hh;
    }
    __syncthreads();
  }
  hn[tid] = h_lds[tid >> 7][tid & 127];
  cn[tid] = c_lds[tid >> 7][tid & 127];
}

// ---------------------------------------------------------------------------
// Kernel 3: flash-style self-attention (Q=K=V=lstm_out, unscaled) + residual
// + FC. One wave per (batch, 16-row tile): grid = (S/16, B), block = 32.
// Q-fragments and the [16,128] context accumulator stay in registers; the
// P tile is transposed D->A layout through LDS for the P@V WMMAs.
// ---------------------------------------------------------------------------
__global__ void attn_fc_kernel(const float* __restrict__ lout,
                               const float* __restrict__ fcw,
                               const float* __restrict__ fcb,
                               float* __restrict__ out) {
  __shared__ float pl[16][17];        // P tile staging (transpose)
  __shared__ float al[16][HH + 1];    // attn_out staging (transpose)

  const int lane = threadIdx.x & 31;
  const int col  = lane & 15;
  const int hi   = lane >> 4;
  const int b    = blockIdx.y;
  const int mt   = blockIdx.x;
  const float* base = lout + (size_t)b * SS * HH;

  // Q fragments: A row = lane&15 of the 16-row tile
  v2f qf[32];
  {
    const float* qrow = base + (size_t)(mt * 16 + col) * HH;
#pragma unroll
    for (int ks = 0; ks < 32; ++ks) {
      const int k = ks * 4 + hi * 2;
      qf[ks].x = qrow[k]; qf[ks].y = qrow[k + 1];
    }
  }

  v8f o[8];                            // context accumulator [16 rows, 128 cols]
#pragma unroll
  for (int nn = 0; nn < 8; ++nn)
#pragma unroll
    for (int v = 0; v < 8; ++v) o[nn][v] = 0.0f;

  float mrow[8], lrow[8];
#pragma unroll
  for (int v = 0; v < 8; ++v) { mrow[v] = -__builtin_inff(); lrow[v] = 0.0f; }

  for (int jt = 0; jt < SS / 16; ++jt) {
    // S_tile = Q @ K^T  (16x16), K-dim = 128
    v8f s;
#pragma unroll
    for (int v = 0; v < 8; ++v) s[v] = 0.0f;
    const float* krow = base + (size_t)(jt * 16 + col) * HH;   // B col = K-row index
#pragma unroll
    for (int ks = 0; ks < 32; ++ks) {
      const int k = ks * 4 + hi * 2;
      v2f bb; bb.x = krow[k]; bb.y = krow[k + 1];
      s = __builtin_amdgcn_wmma_f32_16x16x4_f32(false, qf[ks], false, bb,
                                                (short)0, s, false, false);
    }

    // online softmax (per-row stats; 16-lane butterfly within each half-wave)
#pragma unroll
    for (int v = 0; v < 8; ++v) {
      float mx = s[v];
      for (int off = 1; off < 16; off <<= 1)
        mx = fmaxf(mx, __shfl_xor(mx, off, 16));
      const float mnew = fmaxf(mrow[v], mx);
      const float corr = __expf(mrow[v] - mnew);
      const float p    = __expf(s[v] - mnew);
      float ps = p;
      for (int off = 1; off < 16; off <<= 1)
        ps += __shfl_xor(ps, off, 16);
      lrow[v] = lrow[v] * corr + ps;
      mrow[v] = mnew;
      s[v] = p;
#pragma unroll
      for (int nn = 0; nn < 8; ++nn) o[nn][v] *= corr;
    }

    // transpose P (D layout -> A layout) through LDS, then O += P @ V
    __syncthreads();
#pragma unroll
    for (int v = 0; v < 8; ++v) pl[v + 8 * hi][col] = s[v];
    __syncthreads();
#pragma unroll
    for (int ks = 0; ks < 4; ++ks) {                 // K-dim = 16
      const int k = ks * 4 + hi * 2;
      v2f a; a.x = pl[col][k]; a.y = pl[col][k + 1];
      const float* v0 = base + (size_t)(jt * 16 + k) * HH;
      const float* v1 = base + (size_t)(jt * 16 + k + 1) * HH;
#pragma unroll
      for (int nn = 0; nn < 8; ++nn) {
        v2f bb; bb.x = v0[nn * 16 + col]; bb.y = v1[nn * 16 + col];
        o[nn] = __builtin_amdgcn_wmma_f32_16x16x4_f32(false, a, false, bb,
                                                      (short)0, o[nn], false, false);
      }
    }
    __syncthreads();
  }

  // normalize + residual, stage (transposed) for the FC GEMM
#pragma unroll
  for (int nn = 0; nn < 8; ++nn)
#pragma unroll
    for (int v = 0; v < 8; ++v) {
      const int r = v + 8 * hi;
      const float val = o[nn][v] / lrow[v] +
                        base[(size_t)(mt * 16 + r) * HH + nn * 16 + col];
      al[r][nn * 16 + col] = val;
    }
  __syncthreads();

  // FC: [16,128] @ fc_w^T[128,64] + fc_b  -> 4 N-tiles
#pragma unroll
  for (int nt = 0; nt < 4; ++nt) {
    const float bias = fcb[nt * 16 + col];
    v8f acc;
#pragma unroll
    for (int v = 0; v < 8; ++v) acc[v] = bias;
    const float* wrow = fcw + (size_t)(nt * 16 + col) * HH;
#pragma unroll
    for (int ks = 0; ks < 32; ++ks) {
      const int k = ks * 4 + hi * 2;
      v2f a;  a.x = al[col][k]; a.y = al[col][k + 1];
      v2f bb; bb.x = wrow[k];   bb.y = wrow[k + 1];
      acc = __builtin_amdgcn_wmma_f32_16x16x4_f32(false, a, false, bb,
                                                  (short)0, acc, false, false);
    }
#pragma unroll
    for (int v = 0; v < 8; ++v)
      out[(size_t)((b * SS) + mt * 16 + v + 8 * hi) * OUTD + nt * 16 + col] = acc[v];
  }
}

// ---------------------------------------------------------------------------
extern "C" void kernel_launch(void* const* d_in, const int* in_sizes, int n_in,
                              void* d_out, int out_size, void* d_ws, size_t ws_size,
                              hipStream_t stream) {
  const float* x   = (const float*)d_in[0];
  const float* h0  = (const float*)d_in[1];
  const float* c0  = (const float*)d_in[2];
  const float* Wih = (const float*)d_in[3];
  const float* Whh = (const float*)d_in[4];
  const float* bih = (const float*)d_in[5];
  const float* bhh = (const float*)d_in[6];
  const float* fcw = (const float*)d_in[7];
  const float* fcb = (const float*)d_in[8];
  float* out = (float*)d_out;

  float* xproj = (float*)d_ws;                              // 32 MB
  float* lout  = xproj + (size_t)BB * SS * G4;              //  8 MB

  float* hn = out + (size_t)BB * SS * OUTD;
  float* cn = hn + (size_t)BB * HH;

  xproj_kernel<<<dim3(BB * SS / 16, G4 / 16 / 4), 128, 0, stream>>>(
      x, Wih, bih, bhh, xproj);
  lstm_kernel<<<1, 1024, 0, stream>>>(xproj, h0, c0, Whh, lout, hn, cn);
  attn_fc_kernel<<<dim3(SS / 16, BB), 32, 0, stream>>>(lout, fcw, fcb, out);
}